// GATLayer_30391188587350
// MI455X (gfx1250) — compile-verified
//
#include <hip/hip_runtime.h>
#include <math.h>

typedef __attribute__((ext_vector_type(2))) float v2f;
typedef __attribute__((ext_vector_type(8))) float v8f;

#define HEADS 4
#define OUT_DIM 32
#define FDIM 128   // HEADS*OUT_DIM == INPUT_DIM == 128
#define LRELU 0.2f

// ---------------- helpers: float <-> order-preserving u32 ----------------
__device__ __forceinline__ unsigned f32_ord(float f) {
  unsigned u = __float_as_uint(f);
  return (u & 0x80000000u) ? ~u : (u | 0x80000000u);
}
__device__ __forceinline__ float ord_f32(unsigned u) {
  return (u & 0x80000000u) ? __uint_as_float(u ^ 0x80000000u)
                           : __uint_as_float(~u);
}

// ---------------- init (graph-replay safe: re-zero every call) -----------
__global__ void k_init(float* __restrict__ agg, float* __restrict__ denom,
                       unsigned* __restrict__ gmax, int n_agg, int n_denom) {
  int i = blockIdx.x * blockDim.x + threadIdx.x;
  if (i < n_agg) agg[i] = 0.0f;
  if (i < n_denom) denom[i] = 0.0f;
  if (i == 0) *gmax = 0x007FFFFFu;  // encoded -inf
}

// ---------------- GEMM: H[N,128] = X[N,128] @ W[128,128] via f32 WMMA ----
// One wave computes a 16-row x 128-col stripe.
// W staged in LDS interleaved by K-pairs: Ws[kp][col][j] = W[2*kp+j][col],
// so a B fragment (W[kk][col], W[kk+1][col]) is one aligned ds_load_b64.
__global__ __launch_bounds__(128) void k_gemm_wmma(
    const float* __restrict__ X, const float* __restrict__ Wg,
    float* __restrict__ H, int nrows) {
  __shared__ float Ws[FDIM / 2][FDIM][2];  // 64 KB
  const int tid = threadIdx.x;
  // 64 threads each own one K-pair row; interleave two W rows into LDS.
  if (tid < FDIM / 2) {
    const int kp = tid;
    const float* w0 = Wg + (size_t)(2 * kp) * FDIM;
    const float* w1 = w0 + FDIM;
    for (int c = 0; c < FDIM; c += 4) {
      float4 a = *(const float4*)(w0 + c);
      float4 b = *(const float4*)(w1 + c);
      *(v2f*)&Ws[kp][c + 0][0] = (v2f){a.x, b.x};
      *(v2f*)&Ws[kp][c + 1][0] = (v2f){a.y, b.y};
      *(v2f*)&Ws[kp][c + 2][0] = (v2f){a.z, b.z};
      *(v2f*)&Ws[kp][c + 3][0] = (v2f){a.w, b.w};
    }
  }
  __syncthreads();

  const int wave = tid >> 5;
  const int lane = tid & 31;
  const int row0 = (blockIdx.x * 4 + wave) * 16;
  if (row0 >= nrows) return;

  v8f acc[8];
#pragma unroll
  for (int t = 0; t < 8; ++t) acc[t] = (v8f)(0.0f);

  int mrow = row0 + (lane & 15);
  if (mrow >= nrows) mrow = nrows - 1;  // safe clamp (N % 16 == 0 normally)
  const float* xr = X + (size_t)mrow * FDIM;
  const int khalf = (lane >> 4) << 1;  // lanes 0-15 -> K+0/1, lanes 16-31 -> K+2/3
  const int cl = lane & 15;

  for (int k = 0; k < FDIM; k += 4) {
    const int kk = k + khalf;                 // even
    const v2f a = *(const v2f*)(xr + kk);     // global_load_b64
    const int kp = kk >> 1;
#pragma unroll
    for (int t = 0; t < 8; ++t) {
      const v2f b = *(const v2f*)&Ws[kp][t * 16 + cl][0];  // ds_load_b64
      acc[t] = __builtin_amdgcn_wmma_f32_16x16x4_f32(
          /*neg_a=*/false, a, /*neg_b=*/false, b,
          /*c_mod=*/(short)0, acc[t], /*reuse_a=*/false, /*reuse_b=*/false);
    }
  }

  const int rbase = row0 + ((lane >> 4) ? 8 : 0);
#pragma unroll
  for (int r = 0; r < 8; ++r) {
    if (rbase + r >= nrows) break;
    float* hr = H + (size_t)(rbase + r) * FDIM + cl;
#pragma unroll
    for (int t = 0; t < 8; ++t) hr[t * 16] = acc[t][r];
  }
}

// ---------------- per-node attention scores Hs, Ht -----------------------
__global__ __launch_bounds__(256) void k_scores(
    const float* __restrict__ H, const float* __restrict__ Al,
    const float* __restrict__ Ar, float* __restrict__ Hs,
    float* __restrict__ Ht, int n) {
  __shared__ float sAl[FDIM], sAr[FDIM];
  if (threadIdx.x < FDIM) {
    sAl[threadIdx.x] = Al[threadIdx.x];
    sAr[threadIdx.x] = Ar[threadIdx.x];
  }
  __syncthreads();
  int nidx = blockIdx.x * blockDim.x + threadIdx.x;
  if (nidx >= n) return;
  const float* hr = H + (size_t)nidx * FDIM;
#pragma unroll
  for (int h = 0; h < HEADS; ++h) {
    float s = 0.f, t = 0.f;
#pragma unroll
    for (int d = 0; d < OUT_DIM; ++d) {
      float v = hr[h * OUT_DIM + d];
      s += v * sAl[h * OUT_DIM + d];
      t += v * sAr[h * OUT_DIM + d];
    }
    Hs[nidx * HEADS + h] = s;
    Ht[nidx * HEADS + h] = t;
  }
}

// ---------------- global max of leaky_relu edge scores -------------------
__global__ __launch_bounds__(256) void k_edge_max(
    const int* __restrict__ ei, const float* __restrict__ Hs,
    const float* __restrict__ Ht, unsigned* __restrict__ gmax, int ne) {
  int e = blockIdx.x * blockDim.x + threadIdx.x;
  float m = -3.0e38f;
  if (e < ne) {
    int s = ei[e], t = ei[ne + e];
#pragma unroll
    for (int h = 0; h < HEADS; ++h) {
      float v = Hs[s * HEADS + h] + Ht[t * HEADS + h];
      v = v > 0.f ? v : LRELU * v;
      m = fmaxf(m, v);
    }
  }
#pragma unroll
  for (int off = 16; off > 0; off >>= 1)
    m = fmaxf(m, __shfl_xor(m, off, 32));
  if ((threadIdx.x & 31) == 0) atomicMax(gmax, f32_ord(m));
}

// ---------------- exp(e - max), segment-sum denominators -----------------
__global__ __launch_bounds__(256) void k_edge_exp(
    const int* __restrict__ ei, const float* __restrict__ Hs,
    const float* __restrict__ Ht, const unsigned* __restrict__ gmax,
    float* __restrict__ eexp, float* __restrict__ denom, int ne) {
  int e = blockIdx.x * blockDim.x + threadIdx.x;
  if (e >= ne) return;
  const float mx = ord_f32(*gmax);
  int s = ei[e], t = ei[ne + e];
#pragma unroll
  for (int h = 0; h < HEADS; ++h) {
    float v = Hs[s * HEADS + h] + Ht[t * HEADS + h];
    v = v > 0.f ? v : LRELU * v;
    float ex = expf(v - mx);
    eexp[(size_t)e * HEADS + h] = ex;
    atomicAdd(&denom[s * HEADS + h], ex);
  }
}

// ---------------- weighted scatter-add aggregation -----------------------
// 32 threads per edge; each handles 4 consecutive features (float4 gather).
__global__ __launch_bounds__(256) void k_aggregate(
    const int* __restrict__ ei, const float* __restrict__ H,
    const float* __restrict__ eexp, const float* __restrict__ denom,
    float* __restrict__ agg, int ne) {
  long long gid = (long long)blockIdx.x * blockDim.x + threadIdx.x;
  int e = (int)(gid >> 5);
  if (e >= ne) return;
  int q = (int)(gid & 31);
  int c0 = q * 4;
  int h = q >> 3;  // 32 features per head
  int s = ei[e], t = ei[ne + e];
  float alpha =
      eexp[(size_t)e * HEADS + h] / (denom[s * HEADS + h] + 1e-8f);
  const float4 hv = *(const float4*)(H + (size_t)t * FDIM + c0);
  float* dst = agg + (size_t)s * FDIM + c0;
  atomicAdd(dst + 0, alpha * hv.x);
  atomicAdd(dst + 1, alpha * hv.y);
  atomicAdd(dst + 2, alpha * hv.z);
  atomicAdd(dst + 3, alpha * hv.w);
}

// ---------------- ELU in place -------------------------------------------
__global__ __launch_bounds__(256) void k_elu(float* __restrict__ io, int n) {
  int i = blockIdx.x * blockDim.x + threadIdx.x;
  if (i >= n) return;
  float x = io[i];
  io[i] = x > 0.f ? x : expm1f(x);
}

// ---------------- host-side orchestration --------------------------------
extern "C" void kernel_launch(void* const* d_in, const int* in_sizes, int n_in,
                              void* d_out, int out_size, void* d_ws,
                              size_t ws_size, hipStream_t stream) {
  const float* X = (const float*)d_in[0];
  const int* ei = (const int*)d_in[1];
  const float* W = (const float*)d_in[2];
  const float* Al = (const float*)d_in[3];
  const float* Ar = (const float*)d_in[4];
  const int N = in_sizes[0] / FDIM;
  const int E = in_sizes[1] / 2;

  char* ws = (char*)d_ws;
  size_t off = 0;
  auto carve = [&](size_t bytes) -> char* {
    char* p = ws + off;
    off = (off + bytes + 255) & ~(size_t)255;
    return p;
  };
  float* H = (float*)carve((size_t)N * FDIM * sizeof(float));
  float* Hs = (float*)carve((size_t)N * HEADS * sizeof(float));
  float* Ht = (float*)carve((size_t)N * HEADS * sizeof(float));
  float* denom = (float*)carve((size_t)N * HEADS * sizeof(float));
  unsigned* gmax = (unsigned*)carve(256);
  float* eexp = (float*)carve((size_t)E * HEADS * sizeof(float));
  float* agg = (float*)d_out;

  const int n_agg = N * FDIM;
  k_init<<<(n_agg + 255) / 256, 256, 0, stream>>>(agg, denom, gmax, n_agg,
                                                  N * HEADS);

  const int mtiles = (N + 15) / 16;
  k_gemm_wmma<<<(mtiles + 3) / 4, 128, 0, stream>>>(X, W, H, N);

  k_scores<<<(N + 255) / 256, 256, 0, stream>>>(H, Al, Ar, Hs, Ht, N);

  k_edge_max<<<(E + 255) / 256, 256, 0, stream>>>(ei, Hs, Ht, gmax, E);

  k_edge_exp<<<(E + 255) / 256, 256, 0, stream>>>(ei, Hs, Ht, gmax, eexp,
                                                  denom, E);

  long long aggthreads = (long long)E * 32;
  k_aggregate<<<(unsigned)((aggthreads + 255) / 256), 256, 0, stream>>>(
      ei, H, eexp, denom, agg, E);

  k_elu<<<(n_agg + 255) / 256, 256, 0, stream>>>(agg, n_agg);
}